// ModelNew_3556232922297
// MI455X (gfx1250) — compile-verified
//
#include <hip/hip_runtime.h>

// ---------------------------------------------------------------------------
// Conv3D(16ch->32ch, 3x3x3, VALID) + bias + min-over-depth + channel softmax
// Implicit-GEMM on CDNA5 WMMA: V_WMMA_F32_16X16X4_F32
//   M = output pixels (2 x 16 per wave), N = OC(32) as two 16-wide tiles,
//   K = IC*27 = 432 -> 108 16x16x4 steps. Depth-min fused in registers.
// A loads use uniform SGPR base + i32 VGPR index (GVS + scale_offset);
// B fragments K-paired in LDS and reused across both M-tiles.
// ---------------------------------------------------------------------------

typedef float v2f __attribute__((ext_vector_type(2)));
typedef float v8f __attribute__((ext_vector_type(8)));

#define B_     16
#define IC_    16
#define D_     16
#define H_     128
#define W_     128
#define HW_    (H_*W_)
#define ICDHW_ (IC_*D_*H_*W_)
#define OC_    32
#define DP_    14          // D' = 16-2
#define HP_    126
#define WP_    126
#define P_     (HP_*WP_)   // 15876 output pixels
#define KK_    (IC_*27)    // 432 reduction length
#define KSTEPS_ (KK_/4)    // 108 WMMA k-steps
#define THREADS_ 128
#define WAVES_   4
#define MT_      2                    // M-tiles per wave
#define PIXPW_   (16*MT_)             // 32 pixels per wave
#define PIXPB_   (PIXPW_*WAVES_)      // 128 pixels per block

__launch_bounds__(THREADS_)
__global__ void conv3d_min_softmax_wmma(const float* __restrict__ x,
                                        const float* __restrict__ wgt,
                                        const float* __restrict__ bias,
                                        float* __restrict__ out)
{
    // B staged K-paired: sBw[(k>>1)*64 + oc*2 + (k&1)]; overlaid by softmax
    // staging after the compute phase (guarded by __syncthreads()).
    __shared__ float sBw[KK_ * OC_];                 // 55296 B
    __shared__ __align__(16) int sDelta[KK_];        // 1728 B

    const int tid  = threadIdx.x;
    const int b    = blockIdx.y;
    const int tile = blockIdx.x;

    for (int i = tid; i < KK_ * OC_; i += THREADS_) {
        int k  = i >> 5;          // 0..431
        int oc = i & 31;
        sBw[(k >> 1) * (OC_ * 2) + oc * 2 + (k & 1)] = wgt[oc * KK_ + k];
    }
    for (int i = tid; i < KK_; i += THREADS_) {
        int ic = i / 27, r = i % 27;
        int kd = r / 9,  r2 = r % 9;
        int kh = r2 / 3, kw = r2 % 3;
        sDelta[i] = ic * (D_ * HW_) + kd * HW_ + kh * W_ + kw;
    }
    __syncthreads();

    const int lane  = tid & 31;
    const int wv    = tid >> 5;
    const int n     = lane & 15;               // N index / A-row index
    const int kbase = (lane >> 4) << 1;        // lane-half -> K {0,1} vs {2,3}
    const int mbase = tile * PIXPB_ + wv * PIXPW_;

    // Uniform base pointer (SGPR); all per-lane indexing stays i32 -> GVS mode.
    const float* __restrict__ xB = x + (size_t)b * ICDHW_;

    // Per-lane input offset of each M-tile's pixel (clamped; invalid lanes load
    // in-bounds garbage, results never stored -> EXEC stays all-1s for WMMA).
    int pix[MT_];
    #pragma unroll
    for (int t = 0; t < MT_; ++t) {
        int p  = mbase + t * 16 + n;
        int pc = (p < P_) ? p : (P_ - 1);
        int hh = pc / WP_;
        pix[t] = hh * W_ + (pc - hh * WP_);
    }

    const float INF = __builtin_inff();
    v8f rmin[2 * MT_];            // [mtile*2 + nhalf]
    #pragma unroll
    for (int a = 0; a < 2 * MT_; ++a)
        #pragma unroll
        for (int i = 0; i < 8; ++i) rmin[a][i] = INF;

    // ================= depth loop with fused running min =================
    for (int dz = 0; dz < DP_; ++dz) {
        v8f acc[2 * MT_];
        #pragma unroll
        for (int a = 0; a < 2 * MT_; ++a) acc[a] = (v8f){};

        const int off0 = pix[0] + dz * HW_;
        const int off1 = pix[1] + dz * HW_;

        #pragma unroll 4
        for (int s = 0; s < KSTEPS_; ++s) {
            const int k0 = s * 4 + kbase;                  // even
            const int2 dd = *(const int2*)(sDelta + k0);   // ds_load_b64

            // B: shared by both M-tiles (one float2 per N-half)
            const float* brow = &sBw[(k0 >> 1) * (OC_ * 2)];
            v2f blo = *(const v2f*)(brow + n * 2);
            v2f bhi = *(const v2f*)(brow + (n + 16) * 2);

            // A: two 16x4 fragments, SGPR-base + i32 index loads
            v2f a0, a1;
            a0[0] = xB[off0 + dd.x];
            a0[1] = xB[off0 + dd.y];
            a1[0] = xB[off1 + dd.x];
            a1[1] = xB[off1 + dd.y];

            acc[0] = __builtin_amdgcn_wmma_f32_16x16x4_f32(
                         false, a0, false, blo, (short)0, acc[0], false, false);
            acc[1] = __builtin_amdgcn_wmma_f32_16x16x4_f32(
                         false, a0, false, bhi, (short)0, acc[1], false, false);
            acc[2] = __builtin_amdgcn_wmma_f32_16x16x4_f32(
                         false, a1, false, blo, (short)0, acc[2], false, false);
            acc[3] = __builtin_amdgcn_wmma_f32_16x16x4_f32(
                         false, a1, false, bhi, (short)0, acc[3], false, false);
        }
        #pragma unroll
        for (int a = 0; a < 2 * MT_; ++a)
            #pragma unroll
            for (int i = 0; i < 8; ++i)
                rmin[a][i] = fminf(rmin[a][i], acc[a][i]);
    }

    // ---- bias (depends only on N, commutes with min-over-depth) ----
    const float b_lo = bias[n];
    const float b_hi = bias[n + 16];
    #pragma unroll
    for (int i = 0; i < 8; ++i) {
        rmin[0][i] += b_lo;  rmin[1][i] += b_hi;
        rmin[2][i] += b_lo;  rmin[3][i] += b_hi;
    }

    // ---- overlay softmax staging onto the (now dead) weight LDS ----
    __syncthreads();                       // all waves done reading sBw
    float* sC = sBw;                       // [WAVES_][PIXPW_][OC_] = 16 KB

    // C layout: lane L, vgpr v -> m = v + ((L>=16)?8:0), col = L&15
    const int mofs = (lane >> 4) << 3;
    #pragma unroll
    for (int v = 0; v < 8; ++v) {
        const int m0 = v + mofs;
        float* row0 = sC + (wv * PIXPW_ + m0) * OC_;
        float* row1 = sC + (wv * PIXPW_ + 16 + m0) * OC_;
        row0[n]      = rmin[0][v];
        row0[n + 16] = rmin[1][v];
        row1[n]      = rmin[2][v];
        row1[n + 16] = rmin[3][v];
    }
    // Reads below touch only this wave's rows: same-wave LDS is in-order.

    // ---- per-pixel softmax over 32 channels; every lane owns one pixel ----
    const int pm = mbase + lane;
    if (pm < P_) {
        const float* row = sC + (wv * PIXPW_ + lane) * OC_;
        float vals[OC_];
        float mx = -INF;
        #pragma unroll
        for (int oc = 0; oc < OC_; ++oc) {
            vals[oc] = row[oc];
            mx = fmaxf(mx, vals[oc]);
        }
        float sum = 0.f;
        #pragma unroll
        for (int oc = 0; oc < OC_; ++oc) {
            vals[oc] = __expf(vals[oc] - mx);
            sum += vals[oc];
        }
        const float inv = 1.0f / sum;
        float* op = out + (size_t)b * (OC_ * (size_t)P_) + pm;
        #pragma unroll
        for (int oc = 0; oc < OC_; ++oc)
            op[(size_t)oc * P_] = vals[oc] * inv;
    }
}

extern "C" void kernel_launch(void* const* d_in, const int* in_sizes, int n_in,
                              void* d_out, int out_size, void* d_ws, size_t ws_size,
                              hipStream_t stream) {
    const float* x    = (const float*)d_in[0];  // [16,16,16,128,128] f32
    const float* wgt  = (const float*)d_in[1];  // [32,16,3,3,3] f32
    const float* bias = (const float*)d_in[2];  // [32] f32
    float*       out  = (float*)d_out;          // [16,32,126,126] f32

    dim3 grid((P_ + PIXPB_ - 1) / PIXPB_, B_);  // (125, 16)
    conv3d_min_softmax_wmma<<<grid, THREADS_, 0, stream>>>(x, wgt, bias, out);
}